// PsActivation_20684562497577
// MI455X (gfx1250) — compile-verified
//
#include <hip/hip_runtime.h>
#include <cstdint>

typedef float vfloat4 __attribute__((ext_vector_type(4)));

#define NLEV    1024
#define HCOLS   9
#define THREADS 256
#define BLOCKS  4096

#define NB      8192            // accel buckets over [-4,4); width 1/1024
#define INV_W   1024.0f
#define BW      (1.0f / 1024.0f)
#define XMIN    (-4.0f)

// Shared pool layout (44 KB total):
//   [    0, 36864) : stage  (9216 f32, async-staged h)  -> later ALIASED by
//                    accel  (8192 u32 packed {lo,hi} u16 pairs, 32768 B)
//   [36864, 40960) : grid   (1024 f32, sorted levels = h[:,0])
//   [40960, 45056) : g      (1024 f32, precomputed per-level output)
#define POOL_BYTES 45056
#define GRID_OFF   36864
#define G_OFF      40960

__global__ __launch_bounds__(THREADS) void ps_act_kernel(
    const float* __restrict__ x,    const float* __restrict__ h,
    const float* __restrict__ dvec, const float* __restrict__ Tvec,
    const float* __restrict__ bptr, float* __restrict__ out, int n)
{
    __shared__ __align__(16) unsigned char pool[POOL_BYTES];
    float*    stage = (float*)(pool);            // phases 1-2 only
    uint32_t* accel = (uint32_t*)(pool);         // phase 3+, aliases stage
    float*    grid  = (float*)(pool + GRID_OFF);
    float*    gtab  = (float*)(pool + G_OFF);

    const int tid = threadIdx.x;

    // ---- Phase 1: stage h (36 KB) into LDS via CDNA5 async global->LDS.
    // 9216 floats = 2304 x 16B chunks; 256 threads -> 9 b128 async ops each.
    {
        const uint32_t lbase = (uint32_t)(uintptr_t)&stage[0];
#pragma unroll
        for (int k = 0; k < (NLEV * HCOLS) / (4 * THREADS); ++k) {
            const int      i     = tid + k * THREADS;
            const uint32_t loff  = lbase + (uint32_t)i * 16u;
            const uint64_t gaddr = (uint64_t)(uintptr_t)(h + i * 4);
            asm volatile("global_load_async_to_lds_b128 %0, %1, off"
                         :: "v"(loff), "v"(gaddr) : "memory");
        }
        asm volatile("s_wait_asynccnt 0x0" ::: "memory");
    }
    __syncthreads();

    // Uniform scalars.
    const float d1 = dvec[1], d2 = dvec[2], d3 = dvec[3], d4 = dvec[4];
    const float d5 = dvec[5], d6 = dvec[6], d7 = dvec[7], d8 = dvec[8];
    const float T1 = Tvec[1], T2 = Tvec[2], T3 = Tvec[3], T4 = Tvec[4];
    const float T5 = Tvec[5], T6 = Tvec[6], T7 = Tvec[7], T8 = Tvec[8];
    const float bb = bptr[0];

    // ---- Phase 2: derive grid[] and the per-level output table g[].
    // out(x) depends on x ONLY through nidx: v1 = grid[l], v_t = h[l,t] (t>=2).
#pragma unroll
    for (int r = tid; r < NLEV; r += THREADS) {
        const float* row = stage + r * HCOLS;
        float v = row[0];
        grid[r] = v;
        float acc = (v >= T1) ? d1 : 0.0f;
        v = row[2]; acc += (v >= T2) ? d2 : 0.0f;
        v = row[3]; acc += (v >= T3) ? d3 : 0.0f;
        v = row[4]; acc += (v >= T4) ? d4 : 0.0f;
        v = row[5]; acc += (v >= T5) ? d5 : 0.0f;
        v = row[6]; acc += (v >= T6) ? d6 : 0.0f;
        v = row[7]; acc += (v >= T7) ? d7 : 0.0f;
        v = row[8]; acc += (v >= T8) ? d8 : 0.0f;
        gtab[r] = acc - bb;
    }
    __syncthreads();   // stage is dead from here; accel may overwrite it

    // Branchless capped lower_bound over the 1024 sorted levels.
    auto lbound = [&](float key) -> int {
        int pos = 0;
#pragma unroll
        for (int step = NLEV / 2; step >= 1; step >>= 1)
            if (grid[pos + step - 1] < key) pos += step;
        return pos;    // == min(count(grid < key), 1023)
    };

    // ---- Phase 3: build accel table. Bucket j covers [XMIN+j*BW, XMIN+(j+1)*BW).
    // Window widened by +-1 bucket so float rounding of the bucketize is safe.
    for (int j = tid; j < NB; j += THREADS) {
        const int jm = (j - 1 < 0) ? 0 : (j - 1);
        const int jp = (j + 2 > NB) ? NB : (j + 2);
        const uint32_t lo = (uint32_t)lbound(XMIN + (float)jm * BW);
        const uint32_t hi = (uint32_t)lbound(XMIN + (float)jp * BW);
        accel[j] = lo | (hi << 16);
    }
    __syncthreads();

    // ---- Steady state: ~4 LDS ops per element.
    auto eval1 = [&](float xv) -> float {
        int j = (int)floorf((xv - XMIN) * INV_W);
        j = j < 0 ? 0 : (j > NB - 1 ? NB - 1 : j);
        const uint32_t pr = accel[j];
        int pos = (int)(pr & 0xffffu);
        const int hi = (int)(pr >> 16);
        while (pos < hi && grid[pos] < xv) ++pos;      // expected <1 iter
        const int idx = pos < 1 ? 1 : pos;             // pos <= 1023 by cap
        const float left  = grid[idx - 1];
        const float right = grid[idx];
        const bool  tl    = fabsf(xv - left) < fabsf(xv - right);
        const int   nidx  = tl ? (idx - 1) : idx;
        return gtab[nidx];
    };

    const int gsize = (int)(gridDim.x * blockDim.x);
    const int gid   = (int)(blockIdx.x * blockDim.x) + tid;
    const int nvec  = n >> 2;

    const vfloat4* __restrict__ xin = (const vfloat4*)x;
    vfloat4*       __restrict__ op  = (vfloat4*)out;

    for (int i = gid; i < nvec; i += gsize) {
        vfloat4 xv = __builtin_nontemporal_load(xin + i);
        vfloat4 ov;
        ov.x = eval1(xv.x);
        ov.y = eval1(xv.y);
        ov.z = eval1(xv.z);
        ov.w = eval1(xv.w);
        __builtin_nontemporal_store(ov, op + i);
    }
    for (int i = (nvec << 2) + gid; i < n; i += gsize)
        out[i] = eval1(x[i]);
}

extern "C" void kernel_launch(void* const* d_in, const int* in_sizes, int n_in,
                              void* d_out, int out_size, void* d_ws, size_t ws_size,
                              hipStream_t stream) {
    const float* x  = (const float*)d_in[0];   // (64,512,1024) fp32
    const float* h  = (const float*)d_in[1];   // (1024, 9) fp32
    const float* dv = (const float*)d_in[2];   // (9,) fp32
    const float* Tv = (const float*)d_in[3];   // (9,) fp32
    const float* b  = (const float*)d_in[4];   // scalar fp32
    float* out = (float*)d_out;
    const int n = out_size;                    // 33,554,432

    hipLaunchKernelGGL(ps_act_kernel, dim3(BLOCKS), dim3(THREADS), 0, stream,
                       x, h, dv, Tv, b, out, n);
}